// Echo_61280593380089
// MI455X (gfx1250) — compile-verified
//
#include <hip/hip_runtime.h>
#include <hip/hip_bf16.h>
#include <math.h>

// Problem constants (match reference)
#define BATCH 2048
#define D_MAX 64
#define DIM   512

typedef __attribute__((ext_vector_type(2))) float v2f;
typedef __attribute__((ext_vector_type(4))) float v4f;
typedef __attribute__((ext_vector_type(8))) float v8f;

// ---------------------------------------------------------------------------
// Kernel 0: c[d] = sigmoid(cap_param[d])
// ---------------------------------------------------------------------------
__global__ __launch_bounds__(DIM) void prep_kernel(const float* __restrict__ cap,
                                                   float* __restrict__ c) {
    int d = threadIdx.x;
    float x = cap[d];
    c[d] = 1.0f / (1.0f + expf(-x));
}

// ---------------------------------------------------------------------------
// Kernel 1: noise[b,d] = sum_k c[d]^k * z[nbr[b,k], d]
// One block per batch row b. 128 threads, each owns 4 consecutive dims:
// global_load_b128 per thread per neighbor row (512B contiguous per wave),
// all L2 hits since z_mean (4MB) << 192MB L2. Four independent FMA chains
// per thread give ILP on the serial c^k recurrence.
// ---------------------------------------------------------------------------
__global__ __launch_bounds__(128) void gather_weight_kernel(
    const float* __restrict__ z, const float* __restrict__ c,
    const int* __restrict__ nbr, float* __restrict__ noise) {
    __shared__ int snbr[D_MAX];
    const int b = blockIdx.x;
    const int t = threadIdx.x;
    if (t < D_MAX) snbr[t] = nbr[(size_t)b * D_MAX + t];
    __syncthreads();

    const int d = t * 4;
    const v4f cc = *(const v4f*)(c + d);
    float w0 = 1.0f, w1 = 1.0f, w2 = 1.0f, w3 = 1.0f;  // c^k running weights
    float a0 = 0.0f, a1 = 0.0f, a2 = 0.0f, a3 = 0.0f;

#pragma unroll 8
    for (int k = 0; k < D_MAX; ++k) {
        if (k + 6 < D_MAX) {
            // gfx1250 global_prefetch_b8: pull the upcoming neighbor row
            // toward the WGP while the current FMAs retire.
            __builtin_prefetch(z + (size_t)snbr[k + 6] * DIM + d, 0, 0);
        }
        const v4f v = *(const v4f*)(z + (size_t)snbr[k] * DIM + d);
        a0 = fmaf(w0, v.x, a0);
        a1 = fmaf(w1, v.y, a1);
        a2 = fmaf(w2, v.z, a2);
        a3 = fmaf(w3, v.w, a3);
        w0 *= cc.x;
        w1 *= cc.y;
        w2 *= cc.z;
        w3 *= cc.w;
    }

    v4f r; r.x = a0; r.y = a1; r.z = a2; r.w = a3;
    *(v4f*)(noise + (size_t)b * DIM + d) = r;
}

// ---------------------------------------------------------------------------
// Kernel 2: column sums over the batch via V_WMMA_F32_16X16X4_F32.
// sums[d] = sum_b noise[b,d].  A = ones(16x4) so every row of D carries the
// column sums of the 4 B-rows; accumulate across the batch in the f32 C
// operand.  One block per 16-dim tile (32 blocks), 8 waves each; wave w folds
// rows {w*4 + 32*i .. +3}.  Constant trip count (BATCH/32 = 64) keeps the
// loop control scalar and EXEC all-ones (WMMA requirement).
// ---------------------------------------------------------------------------
__global__ __launch_bounds__(256) void colsum_wmma_kernel(
    const float* __restrict__ noise, float* __restrict__ sums) {
    const int dbase = blockIdx.x * 16;
    const int lane  = threadIdx.x & 31;
    const int wave  = threadIdx.x >> 5;

    v2f a; a.x = 1.0f; a.y = 1.0f;      // all-ones A matrix (16x4)
    v8f acc = {};                        // f32 C/D accumulator

    const int n     = lane & 15;         // output column within tile
    const int khalf = (lane >> 4) * 2;   // rows {0,1} or {2,3} of each 4-group
    const float* base = noise + (size_t)(wave * 4 + khalf) * DIM + dbase + n;

#pragma unroll 4
    for (int i = 0; i < BATCH / 32; ++i) {
        const float* p0 = base + (size_t)i * 32 * DIM;
        v2f bmat;
        bmat.x = p0[0];
        bmat.y = p0[DIM];
        acc = __builtin_amdgcn_wmma_f32_16x16x4_f32(
            /*neg_a=*/false, a, /*neg_b=*/false, bmat,
            /*c_mod=*/(short)0, acc, /*reuse_a=*/false, /*reuse_b=*/false);
    }

    // acc[0] lanes 0..15 hold this wave's partial column sums (row M=0).
    __shared__ float part[8][16];
    if (lane < 16) part[wave][lane] = acc[0];
    __syncthreads();
    if (threadIdx.x < 16) {
        float s = 0.0f;
#pragma unroll
        for (int w = 0; w < 8; ++w) s += part[w][threadIdx.x];
        sums[dbase + threadIdx.x] = s;
    }
}

// ---------------------------------------------------------------------------
// Kernel 3: out[b,d] = z[b,d] + c[d] * (noise[b,d] - sums[d]/BATCH)
// float4 per thread -> b128 loads/stores.
// ---------------------------------------------------------------------------
__global__ __launch_bounds__(256) void combine_kernel(
    const float* __restrict__ z, const float* __restrict__ c,
    const float* __restrict__ noise, const float* __restrict__ sums,
    float* __restrict__ out) {
    const int i4 = (blockIdx.x * 256 + threadIdx.x) * 4;  // over BATCH*DIM
    const int d  = i4 & (DIM - 1);

    const v4f cz = *(const v4f*)(c + d);
    const v4f sm = *(const v4f*)(sums + d);
    const v4f zz = *(const v4f*)(z + i4);
    const v4f nn = *(const v4f*)(noise + i4);

    const float inv = 1.0f / (float)BATCH;
    v4f r;
    r.x = fmaf(cz.x, nn.x - sm.x * inv, zz.x);
    r.y = fmaf(cz.y, nn.y - sm.y * inv, zz.y);
    r.z = fmaf(cz.z, nn.z - sm.z * inv, zz.z);
    r.w = fmaf(cz.w, nn.w - sm.w * inv, zz.w);
    *(v4f*)(out + i4) = r;
}

// ---------------------------------------------------------------------------
// Host launcher
// ---------------------------------------------------------------------------
extern "C" void kernel_launch(void* const* d_in, const int* in_sizes, int n_in,
                              void* d_out, int out_size, void* d_ws, size_t ws_size,
                              hipStream_t stream) {
    const float* z_mean    = (const float*)d_in[0];   // (BATCH, DIM) f32
    const float* cap_param = (const float*)d_in[1];   // (DIM,) f32
    const int*   neighbors = (const int*)d_in[2];     // (BATCH, D_MAX) int

    float* out = (float*)d_out;

    // Workspace layout: c (DIM) | sums (DIM) | noise (BATCH*DIM)
    float* ws_c     = (float*)d_ws;
    float* ws_sums  = ws_c + DIM;
    float* ws_noise = ws_sums + DIM;

    prep_kernel<<<1, DIM, 0, stream>>>(cap_param, ws_c);

    gather_weight_kernel<<<BATCH, 128, 0, stream>>>(z_mean, ws_c, neighbors, ws_noise);

    colsum_wmma_kernel<<<DIM / 16, 256, 0, stream>>>(ws_noise, ws_sums);

    combine_kernel<<<(BATCH * DIM) / (256 * 4), 256, 0, stream>>>(z_mean, ws_c, ws_noise,
                                                                  ws_sums, out);
}